// MoE_74071005987155
// MI455X (gfx1250) — compile-verified
//
#include <hip/hip_runtime.h>
#include <hip/hip_bf16.h>
#include <math.h>

// ---------------------------------------------------------------------------
// MoE router for MI455X (gfx1250):
//   Phase 1: logits = x @ W^T via V_WMMA_F32_16X16X4_F32, double-buffered
//            async global->LDS staging (ASYNCcnt path), fused softmax,
//            transposed prob write [64, 32768] (8 MB, L2-resident).
//   Phase 2: per-expert full bitonic sort of 32768 packed (prob,idx) uint64
//            keys in 256 KB LDS (CDNA5 WGP has 320 KB), emit top-k sorted.
// ---------------------------------------------------------------------------

typedef float v2f __attribute__((ext_vector_type(2)));
typedef float v8f __attribute__((ext_vector_type(8)));
typedef int   v4i_vs __attribute__((__vector_size__(16)));   // matches builtin proto

static constexpr int NUM_T  = 32768;  // tokens (M)
static constexpr int DMODEL = 2048;   // hidden (K)
static constexpr int NE     = 64;     // experts (N)
static constexpr int BM     = 128;    // token rows per block
static constexpr int KC     = 64;     // K chunk staged in LDS
static constexpr int LDX    = 68;     // padded LDS stride (float4-aligned, bank-skewed)
static constexpr int LDP    = 65;     // stride for prob transpose tile

#if defined(__gfx1250__) && __has_builtin(__builtin_amdgcn_global_load_async_to_lds_b128)
#define HAVE_ASYNC_LDS 1
#else
#define HAVE_ASYNC_LDS 0
#endif

#if HAVE_ASYNC_LDS
typedef __attribute__((address_space(1))) v4i_vs* gvec_ptr;  // global (prints __device__)
typedef __attribute__((address_space(3))) v4i_vs* lvec_ptr;  // LDS    (prints __shared__)

__device__ __forceinline__ void async_copy_b128(const float* g, float* l) {
    // GLOBAL_LOAD_ASYNC_TO_LDS_B128: per-lane 16B DMA, tracked by ASYNCcnt.
    __builtin_amdgcn_global_load_async_to_lds_b128(
        (gvec_ptr)(void*)const_cast<float*>(g),
        (lvec_ptr)(void*)l,
        /*imm offset*/0, /*cpol*/0);
}
#if __has_builtin(__builtin_amdgcn_s_wait_asynccnt)
#define WAIT_ASYNC(n) __builtin_amdgcn_s_wait_asynccnt(n)
#else
#define WAIT_ASYNC(n) asm volatile("s_wait_asynccnt %0" :: "i"(n) : "memory")
#endif
#endif

__global__ __launch_bounds__(256)
void router_gemm_softmax(const float* __restrict__ x,
                         const float* __restrict__ W,
                         const float* __restrict__ bias,
                         float* __restrict__ probsT)
{
#if HAVE_ASYNC_LDS
    __shared__ float xs[2][BM * LDX];   // 2 x 34.8 KB
    __shared__ float wsm[2][NE * LDX];  // 2 x 17.4 KB  (104 KB total of 320 KB)
#else
    __shared__ float xs[1][BM * LDX];
    __shared__ float wsm[1][NE * LDX];
#endif

    const int tid  = threadIdx.x;
    const int wave = tid >> 5;       // 8 waves (wave32)
    const int lane = tid & 31;
    const int half = lane >> 4;      // K-pair half per 16x4 f32 A layout
    const int l    = lane & 15;
    const int m0   = wave * 16;      // wave's 16-row stripe in the block
    const long rowBase = (long)blockIdx.x * BM;

    v8f acc[4];
    for (int c = 0; c < 4; ++c)
        for (int i = 0; i < 8; ++i) acc[c][i] = 0.0f;

    const int c4 = (tid & 15) * 4;   // float4 column offset within chunk
    const int r0 = tid >> 4;         // 0..15

    // One K-chunk of WMMAs out of the given LDS buffers.
    auto mma_chunk = [&](const float* xb, const float* wb) {
#pragma unroll 4
        for (int k4 = 0; k4 < KC; k4 += 4) {
            const int ks = k4 + half * 2;
            v2f a;
            a.x = xb[(m0 + l) * LDX + ks];
            a.y = xb[(m0 + l) * LDX + ks + 1];
#pragma unroll
            for (int c = 0; c < 4; ++c) {
                v2f bf;  // B[k][n] = W[n][k]
                bf.x = wb[(c * 16 + l) * LDX + ks];
                bf.y = wb[(c * 16 + l) * LDX + ks + 1];
                acc[c] = __builtin_amdgcn_wmma_f32_16x16x4_f32(
                    false, a, false, bf, (short)0, acc[c], false, false);
            }
        }
    };

#if HAVE_ASYNC_LDS
    // Issue one chunk's async copies: 8 x-tiles + 4 W-tiles per wave (12 ops).
    auto issue = [&](int kc, float* xb, float* wb) {
        for (int p = 0; p < 8; ++p) {
            const int r = p * 16 + r0;
            async_copy_b128(x + (rowBase + r) * DMODEL + kc + c4, xb + r * LDX + c4);
        }
        for (int p = 0; p < 4; ++p) {
            const int n = p * 16 + r0;
            async_copy_b128(W + (long)n * DMODEL + kc + c4, wb + n * LDX + c4);
        }
    };

    constexpr int NCH = DMODEL / KC;  // 32 chunks
    issue(0, xs[0], wsm[0]);
    int buf = 0;
    for (int ci = 0; ci < NCH; ++ci) {
        if (ci + 1 < NCH) {
            issue((ci + 1) * KC, xs[buf ^ 1], wsm[buf ^ 1]);
            WAIT_ASYNC(12);          // oldest 12 (current chunk) complete
        } else {
            WAIT_ASYNC(0);
        }
        __syncthreads();             // buffer visible to all waves
        mma_chunk(xs[buf], wsm[buf]);
        __syncthreads();             // done reading before it is re-filled
        buf ^= 1;
    }
#else
    for (int kc = 0; kc < DMODEL; kc += KC) {
        if (kc + KC < DMODEL)
            __builtin_prefetch(x + (rowBase + r0) * DMODEL + kc + KC + c4, 0, 0);
        for (int p = 0; p < 8; ++p) {
            const int r = p * 16 + r0;
            const float4 v = *(const float4*)(x + (rowBase + r) * DMODEL + kc + c4);
            *(float4*)(xs[0] + r * LDX + c4) = v;
        }
        for (int p = 0; p < 4; ++p) {
            const int n = p * 16 + r0;
            const float4 v = *(const float4*)(W + (long)n * DMODEL + kc + c4);
            *(float4*)(wsm[0] + n * LDX + c4) = v;
        }
        __syncthreads();
        mma_chunk(xs[0], wsm[0]);
        __syncthreads();
    }
#endif

    // Bias per lane-column (b is zeros in this problem, but add anyway).
    float bcol[4];
    for (int c = 0; c < 4; ++c) bcol[c] = bias[c * 16 + l];

    // Softmax epilogue: each row's 64 logits live in acc[0..3][i] across the
    // 16 lanes of one half (C layout: lanes 0-15 -> row m0+i, 16-31 -> m0+i+8).
    float* pt = xs[0];  // reuse as 128 x LDP prob tile (8320 <= 128*LDX floats)
    for (int i = 0; i < 8; ++i) {
        float v[4];
        float vmax = -INFINITY;
        for (int c = 0; c < 4; ++c) { v[c] = acc[c][i] + bcol[c]; vmax = fmaxf(vmax, v[c]); }
        for (int m = 1; m < 16; m <<= 1) vmax = fmaxf(vmax, __shfl_xor(vmax, m, 32));
        float s = 0.0f;
        for (int c = 0; c < 4; ++c) { v[c] = __expf(v[c] - vmax); s += v[c]; }
        for (int m = 1; m < 16; m <<= 1) s += __shfl_xor(s, m, 32);
        const float inv = 1.0f / s;
        const int row = m0 + i + half * 8;
        for (int c = 0; c < 4; ++c) pt[row * LDP + c * 16 + l] = v[c] * inv;
    }
    __syncthreads();

    // Transposed write: probsT[col * NUM_T + token] -- 128B-contiguous per wave
    // store so each expert's column is sequential for the top-k kernel.
    for (int col = wave; col < NE; col += 8)
        for (int rs = 0; rs < BM; rs += 32) {
            const int row = rs + lane;
            probsT[(long)col * NUM_T + rowBase + row] = pt[row * LDP + col];
        }
}

// ---------------------------------------------------------------------------
// Phase 2: one workgroup per expert. Full bitonic sort of 32768 keys held
// entirely in LDS (256 KB of the WGP's 320 KB). Key = probBits<<32 | ~idx:
// probs are positive floats so bit order == numeric order, and ~idx gives
// jax top_k's lower-index-first tie-break under a descending sort.
// ---------------------------------------------------------------------------
__global__ __launch_bounds__(1024)
void expert_topk_sort(const float* __restrict__ probsT,
                      float* __restrict__ vals_out,
                      float* __restrict__ idx_out,
                      int k)
{
    extern __shared__ unsigned long long keys[];  // NUM_T entries = 256 KB
    const int e   = blockIdx.x;
    const int tid = threadIdx.x;
    const float* p = probsT + (long)e * NUM_T;

    for (int j = tid; j < NUM_T; j += 1024) {
        const unsigned int vb = __float_as_uint(p[j]);
        keys[j] = ((unsigned long long)vb << 32) | (0xFFFFFFFFu - (unsigned)j);
    }
    __syncthreads();

    for (int size = 2; size <= NUM_T; size <<= 1) {
        for (int stride = size >> 1; stride > 0; stride >>= 1) {
            for (int t = tid; t < NUM_T / 2; t += 1024) {
                const int low = t & (stride - 1);
                const int i   = ((t ^ low) << 1) | low;  // (t/stride)*2*stride + low
                const int j   = i + stride;
                const bool desc = ((i & size) == 0);     // final pass: all descending
                const unsigned long long a = keys[i], b = keys[j];
                const bool sw = desc ? (a < b) : (a > b);
                if (sw) { keys[i] = b; keys[j] = a; }
            }
            __syncthreads();
        }
    }

    for (int r = tid; r < k; r += 1024) {
        const unsigned long long key = keys[r];
        vals_out[(long)e * k + r] = __uint_as_float((unsigned)(key >> 32));
        idx_out[(long)e * k + r]  = (float)(0xFFFFFFFFu - (unsigned)(key & 0xFFFFFFFFu));
    }
}

extern "C" void kernel_launch(void* const* d_in, const int* in_sizes, int n_in,
                              void* d_out, int out_size, void* d_ws, size_t ws_size,
                              hipStream_t stream)
{
    const float* x    = (const float*)d_in[0];  // [32768, 2048] f32
    const float* W    = (const float*)d_in[1];  // [64, 2048] f32
    const float* bias = (const float*)d_in[2];  // [64] f32
    (void)in_sizes; (void)n_in; (void)ws_size;

    const int k = out_size / (2 * NE);          // vals + idx, per expert

    float* probsT   = (float*)d_ws;             // [64, 32768] f32 = 8 MB scratch
    float* vals_out = (float*)d_out;            // [64, k]
    float* idx_out  = vals_out + (size_t)NE * k;

    router_gemm_softmax<<<NUM_T / BM, 256, 0, stream>>>(x, W, bias, probsT);

    const size_t shmem = (size_t)NUM_T * sizeof(unsigned long long);  // 256 KB LDS
    expert_topk_sort<<<NE, 1024, shmem, stream>>>(probsT, vals_out, idx_out, k);
}